// MGSTGNN_764504179294
// MI455X (gfx1250) — compile-verified
//
#include <hip/hip_runtime.h>
#include <cmath>

// ---------------- problem constants (match reference) ----------------
#define NB   16      // batch B
#define NT   12      // time steps T / OUT_STEPS
#define NN   1024    // nodes N
#define NDIN 3       // input features
#define NH   64      // hidden H
#define NE   16      // node-embedding dim E
#define NGH  256     // GAT hidden
#define NMH  256     // GAT MLP hidden

typedef float v2f __attribute__((ext_vector_type(2)));
typedef float v8f __attribute__((ext_vector_type(8)));

// =====================================================================
// fp32 WMMA GEMM: C = A(MxK, row-major) * B(KxN, row-major).
// One wave per 64x16 C strip (4 M-tiles sharing one B fragment), K
// stepped by 4 via v_wmma_f32_16x16x4_f32.  M multiple of 64, N multiple
// of 16, K multiple of 4.
// Per ISA 7.12.2 (f32):
//   A 16x4 : lane m=L&15, kq=L>>4; vgpr0=A[m][2kq], vgpr1=A[m][2kq+1]
//   B 4x16 : lane n=L&15, kq=L>>4; vgpr0=B[2kq][n], vgpr1=B[2kq+1][n]
//   C 16x16: vgpr v -> row kq*8+v, col L&15
// =====================================================================
__global__ __launch_bounds__(32) void k_wmma_gemm(
    const float* __restrict__ A, const float* __restrict__ B,
    float* __restrict__ C, int K, int lda, int ldb, int ldc,
    long aBatch, long bBatch, long cBatch)
{
    A += (long)blockIdx.z * aBatch;
    B += (long)blockIdx.z * bBatch;
    C += (long)blockIdx.z * cBatch;

    const int lane = threadIdx.x;
    const int m    = lane & 15;
    const int kq   = lane >> 4;          // 0 or 1 -> K pair
    const int row0 = blockIdx.x * 64;    // 4 M-tiles per wave
    const int col0 = blockIdx.y * 16;

    const long astr = (long)16 * lda;    // stride between M-tiles
    const float* arow = A + (long)(row0 + m) * lda + 2 * kq;
    const float* bcol = B + (long)(2 * kq) * ldb + (col0 + m);

    v8f acc0 = {}, acc1 = {}, acc2 = {}, acc3 = {};
    for (int k = 0; k < K; k += 4) {
        v2f b;  b.x  = bcol[0];            b.y  = bcol[ldb];
        v2f a0; a0.x = arow[0];            a0.y = arow[1];
        v2f a1; a1.x = arow[astr];         a1.y = arow[astr + 1];
        v2f a2; a2.x = arow[2 * astr];     a2.y = arow[2 * astr + 1];
        v2f a3; a3.x = arow[3 * astr];     a3.y = arow[3 * astr + 1];
        acc0 = __builtin_amdgcn_wmma_f32_16x16x4_f32(
                   false, a0, false, b, (short)0, acc0, false, false);
        acc1 = __builtin_amdgcn_wmma_f32_16x16x4_f32(
                   false, a1, false, b, (short)0, acc1, false, false);
        acc2 = __builtin_amdgcn_wmma_f32_16x16x4_f32(
                   false, a2, false, b, (short)0, acc2, false, false);
        acc3 = __builtin_amdgcn_wmma_f32_16x16x4_f32(
                   false, a3, false, b, (short)0, acc3, false, false);
        arow += 4;
        bcol += 4 * (long)ldb;
    }

    float* crow = C + (long)(row0 + kq * 8) * ldc + (col0 + m);
#pragma unroll
    for (int v = 0; v < 8; ++v) crow[(long)v * ldc] = acc0[v];
    crow += astr;
#pragma unroll
    for (int v = 0; v < 8; ++v) crow[(long)v * ldc] = acc1[v];
    crow += astr;
#pragma unroll
    for (int v = 0; v < 8; ++v) crow[(long)v * ldc] = acc2[v];
    crow += astr;
#pragma unroll
    for (int v = 0; v < 8; ++v) crow[(long)v * ldc] = acc3[v];
}

// ------------- ne = LayerNorm_E(node_emb + te) -------------
__global__ void k_ln_ne(const float* __restrict__ nemb, const float* __restrict__ te,
                        const float* __restrict__ g, const float* __restrict__ b,
                        float* __restrict__ ne)
{
    int n = blockIdx.x * blockDim.x + threadIdx.x;
    if (n >= NN) return;
    float x[NE], m = 0.f;
#pragma unroll
    for (int e = 0; e < NE; ++e) { x[e] = nemb[n * NE + e] + te[e]; m += x[e]; }
    m *= (1.f / NE);
    float v = 0.f;
#pragma unroll
    for (int e = 0; e < NE; ++e) { float d = x[e] - m; v += d * d; }
    v *= (1.f / NE);
    float rs = rsqrtf(v + 1e-12f);
#pragma unroll
    for (int e = 0; e < NE; ++e) ne[n * NE + e] = (x[e] - m) * rs * g[e] + b[e];
}

// ------------- emb = softmax_row(ne @ ne^T), fused (K=16) -------------
__global__ void k_emb_softmax(const float* __restrict__ ne, float* __restrict__ emb)
{
    __shared__ float sne[NE];
    __shared__ float red[256];
    const int row = blockIdx.x, tid = threadIdx.x;
    if (tid < NE) sne[tid] = ne[row * NE + tid];
    __syncthreads();

    float d[4];
#pragma unroll
    for (int q = 0; q < 4; ++q) {
        int j = tid + q * 256;
        float s = 0.f;
#pragma unroll
        for (int e = 0; e < NE; ++e) s += sne[e] * ne[j * NE + e];
        d[q] = s;
    }
    float mx = fmaxf(fmaxf(d[0], d[1]), fmaxf(d[2], d[3]));
    red[tid] = mx; __syncthreads();
    for (int st = 128; st > 0; st >>= 1) {
        if (tid < st) red[tid] = fmaxf(red[tid], red[tid + st]);
        __syncthreads();
    }
    mx = red[0]; __syncthreads();
    float sum = 0.f;
#pragma unroll
    for (int q = 0; q < 4; ++q) { d[q] = expf(d[q] - mx); sum += d[q]; }
    red[tid] = sum; __syncthreads();
    for (int st = 128; st > 0; st >>= 1) {
        if (tid < st) red[tid] += red[tid + st];
        __syncthreads();
    }
    float inv = 1.f / red[0];
#pragma unroll
    for (int q = 0; q < 4; ++q) emb[row * NN + tid + q * 256] = d[q] * inv;
}

// ------------- xs = concat([x, s (* zmul)]) with zero padding -------------
__global__ void k_concat(const float* __restrict__ A, long aB, int aN, int dx,
                         const float* __restrict__ S, const float* __restrict__ zmul,
                         float* __restrict__ xs, int din, int ldx, int total)
{
    int idx = blockIdx.x * blockDim.x + threadIdx.x;
    if (idx >= total) return;
    int j = idx % ldx;
    int n = (idx / ldx) % NN;
    int b = idx / (ldx * NN);
    float v = 0.f;
    if (j < dx) {
        v = A[(long)b * aB + (long)n * aN + j];
    } else if (j < din) {
        int off = (b * NN + n) * NH + (j - dx);
        v = S[off];
        if (zmul) v *= zmul[off];
    }
    xs[idx] = v;
}

// ------------- per-node GCN output ------------------------------------
// out[b,n,o] = act( sum_{k,i} xg[b,n,k,i] * (sum_d ne[n,d]*Wp[d,k,i,o])
//                   + sum_d ne[n,d]*bp[d,o] )
// One 2-wave block per node; xs/agg rows staged in LDS.
__global__ __launch_bounds__(64) void k_gcn_out(
    const float* __restrict__ xs, const float* __restrict__ agg, int din, int ldx,
    const float* __restrict__ ne, const float* __restrict__ Wp,
    const float* __restrict__ bp, float* __restrict__ out, int act)
{
    __shared__ float sx[NB * 128];
    __shared__ float sa[NB * 128];
    __shared__ float sn[NE];
    const int n = blockIdx.x, tid = threadIdx.x;

    for (int idx = tid; idx < NB * din; idx += 64) {
        int b = idx / din, i = idx % din;
        long goff = ((long)b * NN + n) * ldx + i;
        sx[idx] = xs[goff];
        sa[idx] = agg[goff];
    }
    if (tid < NE) sn[tid] = ne[n * NE + tid];
    __syncthreads();

    const int o = tid;                    // 0..63 output feature
    float bias = 0.f;
#pragma unroll
    for (int d = 0; d < NE; ++d) bias += sn[d] * bp[d * NH + o];
    float acc[NB];
#pragma unroll
    for (int b = 0; b < NB; ++b) acc[b] = bias;

    for (int k = 0; k < 2; ++k) {
        const float* xb = k ? sa : sx;
        for (int i = 0; i < din; ++i) {
            float wv = 0.f;
#pragma unroll
            for (int d = 0; d < NE; ++d)
                wv += sn[d] * Wp[((long)(d * 2 + k) * din + i) * NH + o];
#pragma unroll
            for (int b = 0; b < NB; ++b) acc[b] += xb[b * din + i] * wv;
        }
    }
#pragma unroll
    for (int b = 0; b < NB; ++b) {
        float y = acc[b];
        y = (act == 0) ? (1.f / (1.f + expf(-y))) : tanhf(y);
        out[((long)b * NN + n) * NH + o] = y;
    }
}

// ------------- GRU combine: s = r*s + (1-r)*hc -------------
__global__ void k_gru_combine(const float* __restrict__ r, const float* __restrict__ hc,
                              float* __restrict__ s, int total)
{
    int i = blockIdx.x * blockDim.x + threadIdx.x;
    if (i < total) { float rv = r[i]; s[i] = rv * s[i] + (1.f - rv) * hc[i]; }
}

// ================= GAT kernels (per batch b) =================
__global__ void k_wh(const float* __restrict__ src, const float* __restrict__ W,
                     float* __restrict__ Wh, int b)
{
    int j = blockIdx.x;           // 0..2047 (h = concat([x,x]) -> j % N)
    int o = threadIdx.x;          // 0..255
    int n = j & (NN - 1);
    const float* x = src + (long)b * NT * NN * NDIN + (long)n * NDIN;
    Wh[j * NGH + o] = x[0] * W[o] + x[1] * W[NGH + o] + x[2] * W[2 * NGH + o];
}

__global__ void k_e12(const float* __restrict__ Wh, const float* __restrict__ a,
                      float* __restrict__ e1, float* __restrict__ e2)
{
    int j = blockIdx.x * blockDim.x + threadIdx.x;
    if (j >= 2 * NN) return;
    float s1 = 0.f, s2 = 0.f;
    for (int o = 0; o < NGH; ++o) {
        float w = Wh[j * NGH + o];
        s1 += w * a[o];
        s2 += w * a[NGH + o];
    }
    e1[j] = s1; e2[j] = s2;
}

__device__ __forceinline__ float gat_edge(float a, float e1i, float e2j)
{
    if (a > 0.f) { float t = e1i + e2j; return t > 0.f ? t : 0.2f * t; }
    return -9.0e15f;
}

// column-wise (axis=1) softmax stats: max_i and sumexp_i for each column j
__global__ void k_colstats(const float* __restrict__ adj, const float* __restrict__ e1,
                           const float* __restrict__ e2, float* __restrict__ cmax,
                           float* __restrict__ csum)
{
    __shared__ float red[256];
    const int j = blockIdx.x, tid = threadIdx.x;
    const float ej = e2[j];
    float mx = -3.4e38f;
    for (int i = tid; i < 2 * NN; i += 256)
        mx = fmaxf(mx, gat_edge(adj[(long)i * 2 * NN + j], e1[i], ej));
    red[tid] = mx; __syncthreads();
    for (int st = 128; st > 0; st >>= 1) {
        if (tid < st) red[tid] = fmaxf(red[tid], red[tid + st]);
        __syncthreads();
    }
    mx = red[0]; __syncthreads();
    float s = 0.f;
    for (int i = tid; i < 2 * NN; i += 256)
        s += expf(gat_edge(adj[(long)i * 2 * NN + j], e1[i], ej) - mx);
    red[tid] = s; __syncthreads();
    for (int st = 128; st > 0; st >>= 1) {
        if (tid < st) red[tid] += red[tid + st];
        __syncthreads();
    }
    if (tid == 0) { cmax[j] = mx; csum[j] = red[0]; }
}

__global__ void k_att_fill(const float* __restrict__ adj, const float* __restrict__ e1,
                           const float* __restrict__ e2, const float* __restrict__ cmax,
                           const float* __restrict__ csum, float* __restrict__ att)
{
    int idx = blockIdx.x * blockDim.x + threadIdx.x;
    if (idx >= 2 * NN * 2 * NN) return;
    int j = idx & (2 * NN - 1);
    int i = idx >> 11;
    float v = gat_edge(adj[idx], e1[i], e2[j]);
    att[idx] = expf(v - cmax[j]) / csum[j];
}

// feat = elu(concat([hp[:N], hp[N:]], -1)) for one batch
__global__ void k_feat(const float* __restrict__ hp, float* __restrict__ feat)
{
    int idx = blockIdx.x * blockDim.x + threadIdx.x;    // n*512 + c
    if (idx >= NN * 2 * NGH) return;
    int c = idx & (2 * NGH - 1), n = idx >> 9;
    float v = (c < NGH) ? hp[n * NGH + c] : hp[(n + NN) * NGH + (c - NGH)];
    feat[idx] = v > 0.f ? v : expm1f(v);
}

__global__ void k_bias_gelu(float* __restrict__ y, const float* __restrict__ bias,
                            int ncols, int total)
{
    int i = blockIdx.x * blockDim.x + threadIdx.x;
    if (i >= total) return;
    float x = y[i] + bias[i % ncols];
    y[i] = 0.5f * x * (1.f + erff(x * 0.7071067811865476f));
}

// elu(|x|) == |x|, so fc2 epilogue collapses to bias+abs
__global__ void k_bias_abs(const float* __restrict__ y, const float* __restrict__ bias,
                           float* __restrict__ out, int total)
{
    int i = blockIdx.x * blockDim.x + threadIdx.x;
    if (i >= total) return;
    out[i] = fabsf(y[i] + bias[i & (NH - 1)]);
}

// res0 = LayerNorm_H(s + att0), eps=1e-5
__global__ void k_res0(const float* __restrict__ s, const float* __restrict__ att0,
                       const float* __restrict__ g, const float* __restrict__ b,
                       float* __restrict__ res0)
{
    int r = blockIdx.x * blockDim.x + threadIdx.x;
    if (r >= NB * NN) return;
    const float* ps = s + (long)r * NH;
    const float* pa = att0 + (long)r * NH;
    float x[NH], m = 0.f;
#pragma unroll
    for (int k = 0; k < NH; ++k) { x[k] = ps[k] + pa[k]; m += x[k]; }
    m *= (1.f / NH);
    float v = 0.f;
#pragma unroll
    for (int k = 0; k < NH; ++k) { float d = x[k] - m; v += d * d; }
    v *= (1.f / NH);
    float rs = rsqrtf(v + 1e-5f);
    float* po = res0 + (long)r * NH;
#pragma unroll
    for (int k = 0; k < NH; ++k) po[k] = (x[k] - m) * rs * g[k] + b[k];
}

// final: LN_H(stack([s1, base])), then 1x1 conv over (c,k) -> 12 outputs
__global__ void k_final(const float* __restrict__ s1, const float* __restrict__ base,
                        const float* __restrict__ g, const float* __restrict__ bb,
                        const float* __restrict__ cw, const float* __restrict__ cb,
                        float* __restrict__ out)
{
    int r = blockIdx.x * blockDim.x + threadIdx.x;
    if (r >= NB * NN) return;
    int b = r / NN, n = r % NN;
    const float* src0 = s1 + (long)r * NH;
    const float* src1 = base + (long)r * NH;
    float mean[2], rstd[2];
    for (int c = 0; c < 2; ++c) {
        const float* s = c ? src1 : src0;
        float m = 0.f;
#pragma unroll
        for (int k = 0; k < NH; ++k) m += s[k];
        m *= (1.f / NH);
        float v = 0.f;
#pragma unroll
        for (int k = 0; k < NH; ++k) { float d = s[k] - m; v += d * d; }
        v *= (1.f / NH);
        mean[c] = m; rstd[c] = rsqrtf(v + 1e-12f);
    }
    float acc[NT];
#pragma unroll
    for (int o = 0; o < NT; ++o) acc[o] = cb[o];
    for (int c = 0; c < 2; ++c) {
        const float* s = c ? src1 : src0;
        for (int k = 0; k < NH; ++k) {
            float v = (s[k] - mean[c]) * rstd[c] * g[k] + bb[k];
#pragma unroll
            for (int o = 0; o < NT; ++o) acc[o] += v * cw[(o * 2 + c) * NH + k];
        }
    }
#pragma unroll
    for (int o = 0; o < NT; ++o) out[((long)b * NT + o) * NN + n] = acc[o];
}

// =====================================================================
// Host orchestration
// =====================================================================
struct Gcn { const float *Wp, *bp, *lg, *lb; };

extern "C" void kernel_launch(void* const* d_in, const int* in_sizes, int n_in,
                              void* d_out, int out_size, void* d_ws, size_t ws_size,
                              hipStream_t stream)
{
    (void)in_sizes; (void)n_in; (void)out_size; (void)ws_size;
    const float* IN[40];
    for (int i = 0; i < 40; ++i) IN[i] = (const float*)d_in[i];

    // flattened params (recursive dict order of setup_inputs)
    const float* source   = IN[0];
    const float* adj      = IN[1];
    const float* node_emb = IN[2];
    const float* time_emb = IN[3];
    auto getG = [&](int base) { return Gcn{IN[base], IN[base + 1], IN[base + 2], IN[base + 3]}; };
    Gcn G0[3] = { getG(4),  getG(8),  getG(12) };   // gru0 z,r,u
    Gcn G1[3] = { getG(16), getG(20), getG(24) };   // gru1 z,r,u
    const float* gatW  = IN[28];
    const float* gatA  = IN[29];
    const float* fc1w  = IN[30];
    const float* fc1b  = IN[31];
    const float* fc2w  = IN[32];
    const float* fc2b  = IN[33];
    const float* gn_g  = IN[34];
    const float* gn_b  = IN[35];
    const float* fn_g  = IN[36];
    const float* fn_b  = IN[37];
    const float* convw = IN[38];
    const float* convb = IN[39];

    // ---- workspace carve-up (floats), ~75 MB total ----
    float* wsf = (float*)d_ws;
    size_t off = 0;
    auto alloc = [&](size_t nf) { float* p = wsf + off; off += (nf + 63) & ~(size_t)63; return p; };
    float* ws_s    = alloc((size_t)NB * NN * NH);
    float* ws_s1   = alloc((size_t)NB * NN * NH);
    float* ws_res0 = alloc((size_t)NB * NN * NH);
    float* ws_ne   = alloc((size_t)NN * NE);
    float* ws_emb  = alloc((size_t)NN * NN);
    float* ws_xs   = alloc((size_t)NB * NN * 128);
    float* ws_agg  = alloc((size_t)NB * NN * 128);
    float* ws_z    = alloc((size_t)NB * NN * NH);
    float* ws_r    = alloc((size_t)NB * NN * NH);
    float* ws_hc   = alloc((size_t)NB * NN * NH);
    float* ws_att0 = alloc((size_t)NB * NN * NH);
    float* ws_Wh   = alloc((size_t)2 * NN * NGH);
    float* ws_e1   = alloc(2 * NN);
    float* ws_e2   = alloc(2 * NN);
    float* ws_cmax = alloc(2 * NN);
    float* ws_csum = alloc(2 * NN);
    float* ws_att  = alloc((size_t)2 * NN * 2 * NN);
    float* ws_hp   = alloc((size_t)2 * NN * NGH);
    float* ws_feat = alloc((size_t)NN * 2 * NGH);
    float* ws_fc1  = alloc((size_t)NN * NMH);
    float* ws_fc2  = alloc((size_t)NN * NH);

    const int SNH = NB * NN * NH;

    auto gcnRun = [&](const Gcn& g, const float* te_t, int din, int ldx,
                      float* outbuf, int act) {
        k_ln_ne<<<4, 256, 0, stream>>>(node_emb, te_t, g.lg, g.lb, ws_ne);
        k_emb_softmax<<<NN, 256, 0, stream>>>(ws_ne, ws_emb);
        // agg[b] = emb (NxN) @ xs[b] (N x ldx); wave = 64x16 strip
        k_wmma_gemm<<<dim3(NN / 64, ldx / 16, NB), 32, 0, stream>>>(
            ws_emb, ws_xs, ws_agg, NN, NN, ldx, ldx,
            0L, (long)NN * ldx, (long)NN * ldx);
        k_gcn_out<<<NN, 64, 0, stream>>>(ws_xs, ws_agg, din, ldx, ws_ne,
                                         g.Wp, g.bp, outbuf, act);
    };

    auto gruStep = [&](const float* xbase, long aB, int aN, int dx,
                       float* s, const Gcn* G, const float* te_t) {
        int din = dx + NH;
        int ldx = (din + 15) & ~15;          // 80 for gru0, 128 for gru1
        int total = NB * NN * ldx;
        k_concat<<<(total + 255) / 256, 256, 0, stream>>>(
            xbase, aB, aN, dx, s, nullptr, ws_xs, din, ldx, total);
        gcnRun(G[0], te_t, din, ldx, ws_z, 0);          // z = sigmoid(gcn)
        gcnRun(G[1], te_t, din, ldx, ws_r, 0);          // r = sigmoid(gcn)
        k_concat<<<(total + 255) / 256, 256, 0, stream>>>(
            xbase, aB, aN, dx, s, ws_z, ws_xs, din, ldx, total);
        gcnRun(G[2], te_t, din, ldx, ws_hc, 1);         // hc = tanh(gcn)
        k_gru_combine<<<(SNH + 255) / 256, 256, 0, stream>>>(ws_r, ws_hc, s, SNH);
    };

    hipMemsetAsync(ws_s,  0, (size_t)SNH * sizeof(float), stream);
    hipMemsetAsync(ws_s1, 0, (size_t)SNH * sizeof(float), stream);

    // ---- gru0 step t=0 ----
    gruStep(source, (long)NT * NN * NDIN, NDIN, NDIN, ws_s, G0, time_emb);

    // ---- GAT on source[:,0] (per batch to bound scratch) ----
    for (int b = 0; b < NB; ++b) {
        k_wh<<<2 * NN, 256, 0, stream>>>(source, gatW, ws_Wh, b);
        k_e12<<<(2 * NN + 255) / 256, 256, 0, stream>>>(ws_Wh, gatA, ws_e1, ws_e2);
        k_colstats<<<2 * NN, 256, 0, stream>>>(adj, ws_e1, ws_e2, ws_cmax, ws_csum);
        k_att_fill<<<(2 * NN * 2 * NN) / 256, 256, 0, stream>>>(
            adj, ws_e1, ws_e2, ws_cmax, ws_csum, ws_att);
        // hp = att (2048x2048) @ Wh (2048x256)
        k_wmma_gemm<<<dim3(2 * NN / 64, NGH / 16, 1), 32, 0, stream>>>(
            ws_att, ws_Wh, ws_hp, 2 * NN, 2 * NN, NGH, NGH, 0, 0, 0);
        k_feat<<<(NN * 2 * NGH) / 256, 256, 0, stream>>>(ws_hp, ws_feat);
        // fc1: (1024x512)@(512x256) then gelu
        k_wmma_gemm<<<dim3(NN / 64, NMH / 16, 1), 32, 0, stream>>>(
            ws_feat, fc1w, ws_fc1, 2 * NGH, 2 * NGH, NMH, NMH, 0, 0, 0);
        k_bias_gelu<<<(NN * NMH) / 256, 256, 0, stream>>>(ws_fc1, fc1b, NMH, NN * NMH);
        // fc2: (1024x256)@(256x64) then |.| (elu of abs == abs)
        k_wmma_gemm<<<dim3(NN / 64, NH / 16, 1), 32, 0, stream>>>(
            ws_fc1, fc2w, ws_fc2, NMH, NMH, NH, NH, 0, 0, 0);
        k_bias_abs<<<(NN * NH) / 256, 256, 0, stream>>>(
            ws_fc2, fc2b, ws_att0 + (long)b * NN * NH, NN * NH);
    }

    // res0 = LN(s + att0)
    k_res0<<<(NB * NN + 255) / 256, 256, 0, stream>>>(ws_s, ws_att0, gn_g, gn_b, ws_res0);

    // ---- gru1 step t=0 consumes res0 ----
    gruStep(ws_res0, (long)NN * NH, NH, NH, ws_s1, G1, time_emb);

    // ---- interleaved scans: gru0 produces s_t, gru1 immediately consumes it ----
    for (int t = 1; t < NT; ++t) {
        gruStep(source + (long)t * NN * NDIN, (long)NT * NN * NDIN, NDIN, NDIN,
                ws_s, G0, time_emb + t * NE);
        gruStep(ws_s, (long)NN * NH, NH, NH, ws_s1, G1, time_emb + t * NE);
    }

    // ---- final LN + temporal conv -> (B, 12, N, 1) ----
    k_final<<<(NB * NN + 255) / 256, 256, 0, stream>>>(
        ws_s1, ws_s, fn_g, fn_b, convw, convb, (float*)d_out);
}